// DTW_49572512531210
// MI455X (gfx1250) — compile-verified
//
#include <hip/hip_runtime.h>
#include <hip/hip_bf16.h>

typedef __attribute__((ext_vector_type(2))) float v2f;
typedef __attribute__((ext_vector_type(8))) float v8f;

#define BATCH 128
#define NN 512
#define MM 512
#define DD 64
#define BIGV 1e30f

// ---------------------------------------------------------------------------
// Kernel 0: squared row norms of x (B,N,d) and y (B,M,d)
// grid (B, 2), block 512
// ---------------------------------------------------------------------------
__global__ void dtw_sqnorms(const float* __restrict__ x,
                            const float* __restrict__ y,
                            float* __restrict__ x2,
                            float* __restrict__ y2) {
    const int b   = blockIdx.x;
    const int row = threadIdx.x;           // 0..511
    const float* src;
    float* dst;
    if (blockIdx.y == 0) {
        src = x + ((size_t)b * NN + row) * DD;
        dst = x2 + (size_t)b * NN + row;
    } else {
        src = y + ((size_t)b * MM + row) * DD;
        dst = y2 + (size_t)b * MM + row;
    }
    float s = 0.0f;
#pragma unroll
    for (int k = 0; k < DD; k += 4) {
        float4 v = *(const float4*)(src + k);
        s += v.x * v.x + v.y * v.y + v.z * v.z + v.w * v.w;
    }
    *dst = s;
}

// ---------------------------------------------------------------------------
// Kernel 1: D[b,n,m] = x2[n] + y2[m] - 2 * dot(x[b,n,:], y[b,m,:])
// fp32 WMMA 16x16x4, K-loop of 16. One wave per 16x16 tile.
// grid (N/16, M/16/8, cb), block 256 (8 waves; wave w owns one m-tile)
//
// 32-bit A-matrix 16x4 lane layout (ISA 7.12.2):
//   lanes 0-15 : reg0 = A[M=lane, K=0], reg1 = A[M=lane, K=1]
//   lanes 16-31: reg0 = A[M=lane-16, K=2], reg1 = A[M=lane-16, K=3]
// B (4x16) mirrors with N striped across lanes; C/D: reg r holds
//   M = r (lanes 0-15) / M = r+8 (lanes 16-31), N = lane&15.
// ---------------------------------------------------------------------------
__global__ void dtw_gemm_dist(const float* __restrict__ x,
                              const float* __restrict__ y,
                              const float* __restrict__ x2,
                              const float* __restrict__ y2,
                              float* __restrict__ D,
                              int b0) {
    const int lane = threadIdx.x & 31;
    const int wave = threadIdx.x >> 5;
    const int tm   = blockIdx.x;                 // n-tile (rows of D)
    const int tn   = blockIdx.y * 8 + wave;      // m-tile (cols of D)
    const int b    = b0 + blockIdx.z;

    const int hh  = lane >> 4;                   // 0 or 1
    const int idx = lane & 15;

    const int row_base = tm * 16;
    const int col_base = tn * 16;

    // per-lane base pointers: this lane feeds matrix row/col = idx, with
    // K offset 2*hh inside each K-group of 4.
    const float* xa = x + ((size_t)b * NN + row_base + idx) * DD + 2 * hh;
    const float* yb = y + ((size_t)b * MM + col_base + idx) * DD + 2 * hh;

    v8f c = {};
#pragma unroll
    for (int k = 0; k < DD; k += 4) {
        v2f a  = *(const v2f*)(xa + k);
        v2f bm = *(const v2f*)(yb + k);
        c = __builtin_amdgcn_wmma_f32_16x16x4_f32(
                /*neg_a=*/false, a, /*neg_b=*/false, bm,
                /*c_mod=*/(short)0, c, /*reuse_a=*/false, /*reuse_b=*/false);
    }

    // fuse x2 + y2 - 2*xy and store (coalesced along m)
    const float y2v = y2[(size_t)b * MM + col_base + idx];
    float* drow = D + ((size_t)blockIdx.z * NN) * MM + col_base + idx;
#pragma unroll
    for (int r = 0; r < 8; ++r) {
        const int m_local = r + 8 * hh;
        const float x2v = x2[(size_t)b * NN + row_base + m_local];
        drow[(size_t)(row_base + m_local) * MM] = x2v + y2v - 2.0f * c[r];
    }
}

// ---------------------------------------------------------------------------
// Kernel 2: anti-diagonal wavefront DTW. One block per batch, 512 threads.
// R[i][j] = D[i-1][j-1] + min(R[i-1][j-1], R[i-1][j], R[i][j-1])
//
// Thread tid owns row i = tid+1. For fixed i the D address advances by one
// float per diagonal, so each thread streams its own contiguous row of D and
// prefetches the next step's value into a register one iteration ahead —
// the global-load latency hides behind the current step's compute + barrier.
// The thread's own previous R value stays in a register (rprev); only the
// shifted neighbor values go through LDS. One barrier per diagonal.
// ---------------------------------------------------------------------------
__global__ void dtw_wavefront(const float* __restrict__ D,
                              float* __restrict__ out,
                              int b0) {
    __shared__ float diag[3][NN + 1];
    const int tid = threadIdx.x;               // 0..511
    const int i   = tid + 1;                   // this thread's row, 1..512
    const float* Db   = D + (size_t)blockIdx.x * NN * MM;
    const float* dptr = Db + (size_t)tid * MM; // row i-1 of D; index j-1

    // init all three buffers to BIG; R[0][0] = 0 lives on diag t=0
    for (int k = tid; k <= NN; k += NN) {
        diag[0][k] = BIGV;
        diag[1][k] = BIGV;
        diag[2][k] = BIGV;
    }
    if (tid == 0) diag[0][0] = 0.0f;
    __syncthreads();

    float rprev = BIGV;                        // R[i][1 - i] on diag t=1: BIG
    float dnext = (i == 1) ? dptr[0] : 0.0f;   // D value for step t=2 (j=1)

    for (int t = 2; t <= NN + MM; ++t) {
        const float dcur = dnext;
        // prefetch D for step t+1 (independent of LDS state; issued early)
        const int jn = t + 1 - i;
        dnext = (jn >= 1 && jn <= MM) ? dptr[jn - 1] : 0.0f;

        float* cur      = diag[t % 3];
        const float* p1 = diag[(t - 1) % 3];
        const float* p2 = diag[(t - 2) % 3];

        const int j = t - i;
        const float best = fminf(p2[i - 1], fminf(p1[i - 1], rprev));
        const float v = (j >= 1 && j <= MM) ? (dcur + best) : BIGV;
        cur[i] = v;
        if (tid == 0) cur[0] = BIGV;
        rprev = v;
        __syncthreads();
    }
    // thread owning i = NN computed R[N][M] on the last diagonal
    if (tid == NN - 1) {
        out[b0 + blockIdx.x] = rprev;
    }
}

// ---------------------------------------------------------------------------
// host launcher
// ---------------------------------------------------------------------------
extern "C" void kernel_launch(void* const* d_in, const int* in_sizes, int n_in,
                              void* d_out, int out_size, void* d_ws, size_t ws_size,
                              hipStream_t stream) {
    const float* x = (const float*)d_in[0];
    const float* y = (const float*)d_in[1];
    float* out = (float*)d_out;

    float* x2   = (float*)d_ws;
    float* y2   = x2 + (size_t)BATCH * NN;
    float* Dbuf = y2 + (size_t)BATCH * MM;

    const size_t norm_bytes = (size_t)(BATCH * NN + BATCH * MM) * sizeof(float);
    const size_t per_batch  = (size_t)NN * MM * sizeof(float);

    int chunk = 1;
    if (ws_size > norm_bytes) {
        size_t c = (ws_size - norm_bytes) / per_batch;
        if (c > (size_t)BATCH) c = BATCH;
        if (c >= 1) chunk = (int)c;
    }

    dtw_sqnorms<<<dim3(BATCH, 2), 512, 0, stream>>>(x, y, x2, y2);

    for (int b0 = 0; b0 < BATCH; b0 += chunk) {
        int cb = BATCH - b0;
        if (cb > chunk) cb = chunk;
        dtw_gemm_dist<<<dim3(NN / 16, MM / 16 / 8, cb), 256, 0, stream>>>(
            x, y, x2, y2, Dbuf, b0);
        dtw_wavefront<<<dim3(cb), 512, 0, stream>>>(Dbuf, out, b0);
    }
}